// Aggregator_9105330667541
// MI455X (gfx1250) — compile-verified
//
#include <hip/hip_runtime.h>
#include <hip/hip_bf16.h>

typedef __attribute__((ext_vector_type(2))) float v2f;
typedef __attribute__((ext_vector_type(8))) float v8f;

#define DIM 128
#define NEG_SLOPE 0.01f

// ---------------------------------------------------------------------------
// Kernel 0: zero the N_h accumulator (workspace). float4 stores.
// ---------------------------------------------------------------------------
__global__ void zero_ws_kernel(float4* __restrict__ ws, int n4) {
    int i = blockIdx.x * blockDim.x + threadIdx.x;
    if (i < n4) {
        ws[i] = make_float4(0.f, 0.f, 0.f, 0.f);
    }
}

// ---------------------------------------------------------------------------
// Kernel 1: edge scatter.  One wave32 per edge: each lane moves float4 of the
// source embedding, scales by att[e], and atomically accumulates into
// N_h[dst].  global_load_b128 + 4x global_atomic_add_f32 (no return).
// ---------------------------------------------------------------------------
__global__ __launch_bounds__(256) void edge_scatter_kernel(
    const float* __restrict__ embed,
    const float* __restrict__ att,
    const int*   __restrict__ src,
    const int*   __restrict__ dst,
    float*       __restrict__ nh,
    int n_edges) {
    int edge = blockIdx.x * 8 + (threadIdx.x >> 5);
    if (edge >= n_edges) return;
    int lane = threadIdx.x & 31;

    int s = src[edge];
    int d = dst[edge];
    float a = att[edge];

    const float4* ev = (const float4*)(embed + (size_t)s * DIM);
    float4 v = ev[lane];                      // 4 dwords * 32 lanes = 128 dims

    float* o = nh + (size_t)d * DIM + lane * 4;
    atomicAdd(o + 0, v.x * a);
    atomicAdd(o + 1, v.y * a);
    atomicAdd(o + 2, v.z * a);
    atomicAdd(o + 3, v.w * a);
}

// ---------------------------------------------------------------------------
// Kernel 2: per-node bi-interaction GEMMs with V_WMMA_F32_16X16X4_F32.
//   out = lrelu((e+h) @ W1^T + b1) + lrelu((e*h) @ W2^T + b2)
//
// Workgroup = 32 node rows (two 16-row M-tiles), 8 waves; wave w owns output
// columns [16w, 16w+16) for both GEMMs.  Each B fragment (weight pair) feeds
// FOUR WMMAs (2 M-tiles x 2 GEMMs), halving weight traffic from L2 vs the
// one-tile version.
//
// A fragment (16x4 f32): lane%16 = M, {vgpr, lane>=16} = K:
//   a = { X[m][kk + 2*hi], X[m][kk + 2*hi + 1] }
// B fragment (4x16 f32, B = W^T tile so B[k][n] = W[n][k]):
//   lane%16 = N, {vgpr, lane>=16} = K:
//   b = { W[n0+lm][kk + 2*hi], W[n0+lm][kk + 2*hi + 1] }
// C/D (16x16 f32): vgpr v -> row v + 8*hi, col n0 + lm.
// ---------------------------------------------------------------------------
__global__ __launch_bounds__(256) void node_wmma_kernel(
    const float* __restrict__ embed,
    const float* __restrict__ nh,
    const float* __restrict__ W1,
    const float* __restrict__ bias1,
    const float* __restrict__ W2,
    const float* __restrict__ bias2,
    float*       __restrict__ out,
    int n_nodes) {
    // Row padding 132 floats: row r's element k sits at dword 132r+k, so the
    // 16 lanes of a fragment read hit banks (4r..) -> conflict-free b64 reads.
    // 132 keeps [r][even k] 8-byte aligned.
    __shared__ float xs1[32][132];
    __shared__ float xs2[32][132];

    const int row0 = blockIdx.x * 32;
    const int tid  = threadIdx.x;

    // Stage X1 = e + h and X2 = e * h for this 32-row tile into LDS.
    // Clamp rows so the tail block stays in-bounds (garbage rows are never
    // stored).
    for (int i = tid; i < 32 * DIM; i += 256) {
        int r = i >> 7;
        int c = i & (DIM - 1);
        int rr = row0 + r;
        rr = (rr < n_nodes) ? rr : (n_nodes - 1);
        float e = embed[(size_t)rr * DIM + c];
        float h = nh   [(size_t)rr * DIM + c];
        xs1[r][c] = e + h;
        xs2[r][c] = e * h;
    }
    __syncthreads();

    const int lane = tid & 31;
    const int lm   = lane & 15;       // M for A, N for B/C/D
    const int hi   = lane >> 4;       // K sub-offset selector
    const int n0   = (tid >> 5) * 16; // this wave's output-column tile

    const float* w1row = W1 + (size_t)(n0 + lm) * DIM;
    const float* w2row = W2 + (size_t)(n0 + lm) * DIM;

    v8f c1t0 = {}, c1t1 = {};   // X1 @ W1^T, M-tiles 0 and 1
    v8f c2t0 = {}, c2t1 = {};   // X2 @ W2^T, M-tiles 0 and 1

    #pragma unroll 4
    for (int kk = 0; kk < DIM; kk += 4) {
        const int k = kk + 2 * hi;

        v2f bf1 = *(const v2f*)&w1row[k];
        v2f bf2 = *(const v2f*)&w2row[k];

        v2f a1t0 = *(const v2f*)&xs1[lm][k];
        v2f a1t1 = *(const v2f*)&xs1[16 + lm][k];
        v2f a2t0 = *(const v2f*)&xs2[lm][k];
        v2f a2t1 = *(const v2f*)&xs2[16 + lm][k];

        c1t0 = __builtin_amdgcn_wmma_f32_16x16x4_f32(
                   false, a1t0, false, bf1, (short)0, c1t0, false, false);
        c1t1 = __builtin_amdgcn_wmma_f32_16x16x4_f32(
                   false, a1t1, false, bf1, (short)0, c1t1, false, false);
        c2t0 = __builtin_amdgcn_wmma_f32_16x16x4_f32(
                   false, a2t0, false, bf2, (short)0, c2t0, false, false);
        c2t1 = __builtin_amdgcn_wmma_f32_16x16x4_f32(
                   false, a2t1, false, bf2, (short)0, c2t1, false, false);
    }

    const float bb1 = bias1[n0 + lm];
    const float bb2 = bias2[n0 + lm];

    // M-tile 0 (always valid: row0 < n_nodes for every launched block)
    {
        float* ocol = out + (size_t)row0 * DIM + n0 + lm;
        #pragma unroll
        for (int v = 0; v < 8; ++v) {
            int m = v + 8 * hi;
            float o1 = c1t0[v] + bb1;
            o1 = (o1 >= 0.f) ? o1 : NEG_SLOPE * o1;
            float o2 = c2t0[v] + bb2;
            o2 = (o2 >= 0.f) ? o2 : NEG_SLOPE * o2;
            ocol[(size_t)m * DIM] = o1 + o2;
        }
    }
    // M-tile 1 (wave-uniform guard for the tail block)
    if (row0 + 16 < n_nodes) {
        float* ocol = out + (size_t)(row0 + 16) * DIM + n0 + lm;
        #pragma unroll
        for (int v = 0; v < 8; ++v) {
            int m = v + 8 * hi;
            float o1 = c1t1[v] + bb1;
            o1 = (o1 >= 0.f) ? o1 : NEG_SLOPE * o1;
            float o2 = c2t1[v] + bb2;
            o2 = (o2 >= 0.f) ? o2 : NEG_SLOPE * o2;
            ocol[(size_t)m * DIM] = o1 + o2;
        }
    }
}

// ---------------------------------------------------------------------------
// Host-side launcher.
// Input order: entity_embed, att, W1, b1, W2, b2, src, dst
// ---------------------------------------------------------------------------
extern "C" void kernel_launch(void* const* d_in, const int* in_sizes, int n_in,
                              void* d_out, int out_size, void* d_ws, size_t ws_size,
                              hipStream_t stream) {
    const float* embed = (const float*)d_in[0];
    const float* att   = (const float*)d_in[1];
    const float* W1    = (const float*)d_in[2];
    const float* b1    = (const float*)d_in[3];
    const float* W2    = (const float*)d_in[4];
    const float* b2    = (const float*)d_in[5];
    const int*   src   = (const int*)d_in[6];
    const int*   dst   = (const int*)d_in[7];

    const int n_nodes = in_sizes[0] / DIM;   // 50000
    const int n_edges = in_sizes[6];         // 800000

    float* nh = (float*)d_ws;                // [n_nodes, 128] accumulator

    // 1) zero N_h
    {
        int n4 = n_nodes * DIM / 4;
        int blocks = (n4 + 255) / 256;
        zero_ws_kernel<<<blocks, 256, 0, stream>>>((float4*)nh, n4);
    }
    // 2) edge scatter (8 edges per 256-thread block, one wave32 per edge)
    {
        int blocks = (n_edges + 7) / 8;
        edge_scatter_kernel<<<blocks, 256, 0, stream>>>(embed, att, src, dst,
                                                        nh, n_edges);
    }
    // 3) WMMA node kernel (32 rows per block, tail-guarded)
    {
        int blocks = (n_nodes + 31) / 32;
        node_wmma_kernel<<<blocks, 256, 0, stream>>>(embed, nh, W1, b1, W2, b2,
                                                     (float*)d_out, n_nodes);
    }
}